// GraphEncoder_2911987826702
// MI455X (gfx1250) — compile-verified
//
#include <hip/hip_runtime.h>

// ---------------------------------------------------------------------------
// GCN forward on MI455X (gfx1250, wave32).
//  * GEMMs: V_WMMA_F32_16X16X4_F32 (full f32; problem is L2-bound so the f32
//    matrix path costs nothing vs bf16).
//  * A-tiles staged with GLOBAL_LOAD_ASYNC_TO_LDS_B128 (+ s_wait_asynccnt).
//  * Fused epilogues: layer GEMM also emits agg = m*dinv^2 (self-loop term);
//    relu(agg + b) is applied as a pre-op on the next GEMM's LDS tile, so the
//    intermediate h is never materialized after the input projection.
//  * Edge propagation: one wave per edge, float2 per lane, non-returning
//    global_atomic_add_f32 into L2-resident agg.
// ---------------------------------------------------------------------------

typedef __attribute__((ext_vector_type(2))) float v2f;
typedef __attribute__((ext_vector_type(8))) float v8f;

#define HDIM 64

// ---------------- degree / normalization ----------------
__global__ void deg_init_kernel(float* __restrict__ deg, int n) {
    int i = blockIdx.x * blockDim.x + threadIdx.x;
    if (i < n) deg[i] = 1.0f;  // self-loop contributes 1 to in-degree
}

__global__ void deg_count_kernel(const int* __restrict__ dst,
                                 float* __restrict__ deg, int e) {
    int i = blockIdx.x * blockDim.x + threadIdx.x;
    if (i < e) atomicAdd(&deg[dst[i]], 1.0f);
}

__global__ void deg_finalize_kernel(float* __restrict__ deg, int n) {
    int i = blockIdx.x * blockDim.x + threadIdx.x;
    if (i < n) {
        float d = deg[i];
        deg[i] = (d > 0.0f) ? rsqrtf(d) : 0.0f;  // now holds dinv
    }
}

// ---------------- WMMA GEMM with fused pre/post ops ----------------
// out[N,64] = f(A)[N,K] @ W[K,64], where f = relu(.+pre_bias) if pre_bias.
// If agg_out: also writes agg_out = (A@W) * dinv[row]^2  (self-loop init).
// Block = 128 threads = 4 wave32; blockIdx.x = 16-row tile; wave w owns the
// 16-column slab [16w,16w+16). Fragments follow CDNA5 ISA 7.12.2 layouts.
__global__ __launch_bounds__(128)
void gemm_n64_wmma_kernel(const float* __restrict__ A,
                          const float* __restrict__ W,
                          const float* __restrict__ pre_bias,   // nullable
                          const float* __restrict__ post_bias,  // nullable
                          const float* __restrict__ dinv,       // with agg_out
                          float* __restrict__ out,
                          float* __restrict__ agg_out,          // nullable
                          int nrows, int K, int post_relu) {
    __shared__ float As[16 * (128 + 4)];
    const int KP   = K + 4;              // row pitch: 16B-aligned, banks +4/row
    const int tid  = threadIdx.x;
    const int tile = blockIdx.x;

    // ---- async stage of the 16xK A tile: memory -> LDS, no VGPR round trip.
    // OOB tail rows are clamped to a valid row; their outputs are masked at
    // store time and A rows only ever feed their own output row.
    const int nv4 = (16 * K) >> 2;
    for (int i4 = tid; i4 < nv4; i4 += 128) {
        int flat = i4 << 2;
        int r    = flat / K;
        int kk   = flat - r * K;
        int row  = tile * 16 + r;
        if (row >= nrows) row = nrows - 1;
        unsigned ldsoff = (unsigned)((r * KP + kk) * sizeof(float));
        unsigned long long gaddr =
            (unsigned long long)(uintptr_t)(A + (size_t)row * K + kk);
        asm volatile("global_load_async_to_lds_b128 %0, %1, off"
                     :: "v"(ldsoff), "v"(gaddr) : "memory");
    }
    asm volatile("s_wait_asynccnt 0x0" ::: "memory");
    __syncthreads();

    // ---- fused pre-op on the tile: A' = relu(A + pre_bias[k])
    if (pre_bias) {
        for (int i = tid; i < 16 * K; i += 128) {
            int r = i / K, kk = i - r * K;
            float* p = &As[r * KP + kk];
            *p = fmaxf(*p + pre_bias[kk], 0.0f);
        }
        __syncthreads();
    }

    const int wave = tid >> 5;
    const int lane = tid & 31;
    const int m    = lane & 15;          // A row within tile
    const int nl   = lane & 15;          // B/D column within wave's slab
    const int kh   = (lane >> 4) << 1;   // K sub-offset: 0 (lo half) / 2 (hi)
    const int cn   = wave << 4;          // column slab base

    v8f acc = {};
    const float* Wc = W + cn + nl;
    for (int k0 = 0; k0 < K; k0 += 4) {
        int ka = k0 + kh;
        v2f a, b;
        a.x = As[m * KP + ka];
        a.y = As[m * KP + ka + 1];
        b.x = Wc[(size_t)ka * HDIM];
        b.y = Wc[(size_t)(ka + 1) * HDIM];
        acc = __builtin_amdgcn_wmma_f32_16x16x4_f32(
            false, a, false, b, (short)0, acc, false, false);
    }

    float bv = post_bias ? post_bias[cn + nl] : 0.0f;
#pragma unroll
    for (int i = 0; i < 8; ++i) {
        int row = tile * 16 + i + ((lane >> 4) << 3);
        if (row < nrows) {
            float v = acc[i];
            float o = v + bv;
            if (post_relu) o = fmaxf(o, 0.0f);
            size_t idx = (size_t)row * HDIM + cn + nl;
            out[idx] = o;
            if (agg_out) {                      // fused self-loop init
                float di = dinv[row];
                agg_out[idx] = v * di * di;
            }
        }
    }
}

// ---------------- edge scatter: agg[dst] += m[src] * dinv[s]*dinv[t] -------
__global__ __launch_bounds__(256)
void scatter_edges_kernel(const int* __restrict__ src,
                          const int* __restrict__ dst,
                          const float* __restrict__ dinv,
                          const float* __restrict__ m,
                          float* __restrict__ agg, int e) {
    int wave = threadIdx.x >> 5;
    int lane = threadIdx.x & 31;
    int edge = blockIdx.x * 8 + wave;
    if (edge >= e) return;
    int s = src[edge];
    int t = dst[edge];
    float w = dinv[s] * dinv[t];
    float2 v = *(const float2*)(m + (size_t)s * HDIM + 2 * lane);
    float* arow = agg + (size_t)t * HDIM + 2 * lane;
    atomicAdd(arow + 0, v.x * w);
    atomicAdd(arow + 1, v.y * w);
}

// ---------------------------------------------------------------------------
extern "C" void kernel_launch(void* const* d_in, const int* in_sizes, int n_in,
                              void* d_out, int out_size, void* d_ws, size_t ws_size,
                              hipStream_t stream) {
    const float* x     = (const float*)d_in[0];   // [N,128]
    const int*   ei    = (const int*)  d_in[1];   // [2,E]
    const float* w_in  = (const float*)d_in[2];   // [128,64]
    const float* b_in  = (const float*)d_in[3];   // [64]
    const float* wsL   = (const float*)d_in[4];   // [3,64,64]
    const float* bsL   = (const float*)d_in[5];   // [3,64]
    const float* w_out = (const float*)d_in[6];   // [64,64]
    const float* b_out = (const float*)d_in[7];   // [64]

    const int N = in_sizes[0] / 128;
    const int E = in_sizes[1] / 2;
    const int* src = ei;
    const int* dst = ei + E;

    // scratch (floats): dinv | bufA | m | bufB.  h lives in bufA only between
    // the input projection and layer 0; agg ping-pongs bufB/bufA/bufB.
    float* wsf = (float*)d_ws;
    size_t NPAD   = ((size_t)N + 255) & ~(size_t)255;
    float* f_dinv = wsf;
    float* f_bufA = wsf + NPAD;               // h, then agg for layers 1
    float* f_m    = f_bufA + NPAD * HDIM;
    float* f_bufB = f_m + NPAD * HDIM;        // agg for layers 0 and 2

    const int tiles = (N + 15) / 16;

    // symmetric normalization coefficients
    deg_init_kernel    <<<(N + 255) / 256, 256, 0, stream>>>(f_dinv, N);
    deg_count_kernel   <<<(E + 255) / 256, 256, 0, stream>>>(dst, f_dinv, E);
    deg_finalize_kernel<<<(N + 255) / 256, 256, 0, stream>>>(f_dinv, N);

    // input projection: h = relu(x @ w_in + b_in)            -> bufA
    gemm_n64_wmma_kernel<<<tiles, 128, 0, stream>>>(
        x, w_in, nullptr, b_in, nullptr, f_bufA, nullptr, N, 128, 1);

    // layer 0: m = h @ ws0 ; agg(bufB) = m*dinv^2 ; scatter edges
    gemm_n64_wmma_kernel<<<tiles, 128, 0, stream>>>(
        f_bufA, wsL + 0 * HDIM * HDIM, nullptr, nullptr, f_dinv,
        f_m, f_bufB, N, HDIM, 0);
    scatter_edges_kernel<<<(E + 7) / 8, 256, 0, stream>>>(
        src, dst, f_dinv, f_m, f_bufB, E);

    // layer 1: m = relu(aggB + bs0) @ ws1 ; agg(bufA) ; scatter
    gemm_n64_wmma_kernel<<<tiles, 128, 0, stream>>>(
        f_bufB, wsL + 1 * HDIM * HDIM, bsL + 0 * HDIM, nullptr, f_dinv,
        f_m, f_bufA, N, HDIM, 0);
    scatter_edges_kernel<<<(E + 7) / 8, 256, 0, stream>>>(
        src, dst, f_dinv, f_m, f_bufA, E);

    // layer 2: m = relu(aggA + bs1) @ ws2 ; agg(bufB) ; scatter
    gemm_n64_wmma_kernel<<<tiles, 128, 0, stream>>>(
        f_bufA, wsL + 2 * HDIM * HDIM, bsL + 1 * HDIM, nullptr, f_dinv,
        f_m, f_bufB, N, HDIM, 0);
    scatter_edges_kernel<<<(E + 7) / 8, 256, 0, stream>>>(
        src, dst, f_dinv, f_m, f_bufB, E);

    // output: out = relu(aggB + bs2) @ w_out + b_out         -> d_out
    gemm_n64_wmma_kernel<<<tiles, 128, 0, stream>>>(
        f_bufB, w_out, bsL + 2 * HDIM, b_out, nullptr,
        (float*)d_out, nullptr, N, HDIM, 0);
}